// Encoder_47330539602647
// MI455X (gfx1250) — compile-verified
//
#include <hip/hip_runtime.h>

typedef __attribute__((ext_vector_type(2))) float v2f;
typedef __attribute__((ext_vector_type(4))) float v4f;
typedef __attribute__((ext_vector_type(8))) float v8f;

#define C_IN  256
#define C_OUT 128
#define WT_STRIDE 258   // padded K-stride (floats) to avoid LDS bank conflicts

// ---------------------------------------------------------------------------
// 1) degree init: deg[n] = 1 (self loop)
__global__ __launch_bounds__(256) void gcn_deg_init(float* __restrict__ deg, int N) {
    int i = blockIdx.x * blockDim.x + threadIdx.x;
    if (i < N) deg[i] = 1.0f;
}

// 2) degree accumulate over edges at dst
__global__ __launch_bounds__(256) void gcn_deg_accum(const int* __restrict__ dst,
                                                     float* __restrict__ deg, int E) {
    int e = blockIdx.x * blockDim.x + threadIdx.x;
    if (e < E) atomicAdd(&deg[dst[e]], 1.0f);
}

// 3) deg -> rsqrt(deg) in place (deg >= 1 always)
__global__ __launch_bounds__(256) void gcn_dinv(float* __restrict__ deg, int N) {
    int i = blockIdx.x * blockDim.x + threadIdx.x;
    if (i < N) deg[i] = rsqrtf(deg[i]);
}

// ---------------------------------------------------------------------------
// 4) xw = x @ W with V_WMMA_F32_16X16X4_F32.
//    8 waves/block; each wave: 16-row strip x all 128 out-cols (8 N-tiles).
//    Inner loop: batch-load 8 B fragments (one dscnt wait), prefetch next A
//    fragment during the 8 back-to-back WMMAs.
__global__ __launch_bounds__(256) void gcn_gemm_wmma(const float* __restrict__ x,
                                                     const float* __restrict__ W,
                                                     float* __restrict__ xw, int N) {
    extern __shared__ float Wt[];  // [C_OUT][WT_STRIDE] : Wt[c*WT_STRIDE + k] = W[k][c]
    const int tid = threadIdx.x;

    // cooperative transpose-stage of W (coalesced global reads)
    for (int idx = tid; idx < C_IN * C_OUT; idx += 256) {
        int k = idx >> 7;       // 0..255
        int c = idx & 127;      // 0..127
        Wt[c * WT_STRIDE + k] = W[idx];
    }
    __syncthreads();

    const int wave = tid >> 5;
    const int lane = tid & 31;
    const int row_base = (blockIdx.x * 8 + wave) * 16;
    if (row_base >= N) return;                 // wave-uniform: EXEC stays all-1s

    const int m    = lane & 15;                // M index within tile
    const int kh   = (lane >> 4) << 1;         // 0 (lanes 0-15) or 2 (lanes 16-31)
    const int ncol = lane & 15;                // N index within tile
    const int row  = min(row_base + m, N - 1);

    v8f acc[8] = {};                           // 16x128 fp32 accumulators (64 VGPRs)

    const float* xrow  = x + (size_t)row * C_IN + kh;
    const float* wbase = &Wt[ncol * WT_STRIDE + kh];

    v2f a = *(const v2f*)(xrow);               // k = 0 fragment

    #pragma unroll 4
    for (int k = 0; k < C_IN - 4; k += 4) {
        // batch all 8 B fragments -> one s_wait_dscnt covers the group
        v2f b[8];
        #pragma unroll
        for (int t = 0; t < 8; ++t)
            b[t] = *(const v2f*)(wbase + k + t * (16 * WT_STRIDE));

        v2f a_cur = a;
        a = *(const v2f*)(xrow + k + 4);       // prefetch next A (unconditional)

        #pragma unroll
        for (int t = 0; t < 8; ++t)
            acc[t] = __builtin_amdgcn_wmma_f32_16x16x4_f32(
                false, a_cur, false, b[t], (short)0, acc[t], false, false);
    }
    {   // peeled last k-step (k = C_IN-4), no prefetch
        const int k = C_IN - 4;
        v2f b[8];
        #pragma unroll
        for (int t = 0; t < 8; ++t)
            b[t] = *(const v2f*)(wbase + k + t * (16 * WT_STRIDE));
        #pragma unroll
        for (int t = 0; t < 8; ++t)
            acc[t] = __builtin_amdgcn_wmma_f32_16x16x4_f32(
                false, a, false, b[t], (short)0, acc[t], false, false);
    }

    // D layout: VGPR j holds M=j (lanes 0-15) and M=j+8 (lanes 16-31), N=lane&15
    const int rhalf = (lane >> 4) << 3;
    #pragma unroll
    for (int t = 0; t < 8; ++t) {
        #pragma unroll
        for (int j = 0; j < 8; ++j) {
            int r = row_base + rhalf + j;
            if (r < N) xw[(size_t)r * C_OUT + t * 16 + ncol] = acc[t][j];
        }
    }
}

// ---------------------------------------------------------------------------
// 5) out = xw * dinv^2 (self loop) + bias
__global__ __launch_bounds__(256) void gcn_init_out(const float* __restrict__ xw,
                                                    const float* __restrict__ dinv,
                                                    const float* __restrict__ bias,
                                                    float* __restrict__ out, int total) {
    int idx = blockIdx.x * blockDim.x + threadIdx.x;
    if (idx >= total) return;
    int n = idx >> 7;
    int c = idx & 127;
    float di = dinv[n];
    out[idx] = xw[idx] * di * di + bias[c];
}

// 6) edge scatter: one wave per edge; lane handles 4 channels (float4 gather)
__global__ __launch_bounds__(256) void gcn_edge_scatter(const int* __restrict__ src,
                                                        const int* __restrict__ dst,
                                                        const float* __restrict__ xw,
                                                        const float* __restrict__ dinv,
                                                        float* __restrict__ out, int E) {
    int gid = blockIdx.x * blockDim.x + threadIdx.x;
    int e = gid >> 5;
    if (e >= E) return;
    int lane = gid & 31;
    int s = src[e];
    int d = dst[e];
    float norm = dinv[s] * dinv[d];
    v4f v = *(const v4f*)(xw + (size_t)s * C_OUT + lane * 4);
    float* o = out + (size_t)d * C_OUT + lane * 4;
    atomicAdd(o + 0, v.x * norm);
    atomicAdd(o + 1, v.y * norm);
    atomicAdd(o + 2, v.z * norm);
    atomicAdd(o + 3, v.w * norm);
}

// 7) in-place per-channel PReLU
__global__ __launch_bounds__(256) void gcn_prelu(float* __restrict__ out,
                                                 const float* __restrict__ alpha, int total) {
    int idx = blockIdx.x * blockDim.x + threadIdx.x;
    if (idx >= total) return;
    float v = out[idx];
    out[idx] = v > 0.0f ? v : alpha[idx & 127] * v;
}

// ---------------------------------------------------------------------------
extern "C" void kernel_launch(void* const* d_in, const int* in_sizes, int n_in,
                              void* d_out, int out_size, void* d_ws, size_t ws_size,
                              hipStream_t stream) {
    const float* x     = (const float*)d_in[0];
    const int*   ei    = (const int*)d_in[1];   // [2, E] flat: row 0 = src, row 1 = dst
    const float* W     = (const float*)d_in[2];
    const float* bias  = (const float*)d_in[3];
    const float* alpha = (const float*)d_in[4];

    const int N = in_sizes[0] / C_IN;
    const int E = in_sizes[1] / 2;
    const int* src = ei;
    const int* dst = ei + E;

    // workspace layout: xw [N*128 f32] | deg/dinv [N f32]
    float* xw  = (float*)d_ws;
    float* deg = (float*)((char*)d_ws + (size_t)N * C_OUT * sizeof(float));
    float* out = (float*)d_out;

    const int total = N * C_OUT;

    gcn_deg_init<<<(N + 255) / 256, 256, 0, stream>>>(deg, N);
    gcn_deg_accum<<<(E + 255) / 256, 256, 0, stream>>>(dst, deg, E);
    gcn_dinv<<<(N + 255) / 256, 256, 0, stream>>>(deg, N);

    const size_t lds_bytes = (size_t)C_OUT * WT_STRIDE * sizeof(float);  // ~132 KB, fits 320 KB WGP LDS
    gcn_gemm_wmma<<<(N + 127) / 128, 256, lds_bytes, stream>>>(x, W, xw, N);

    gcn_init_out<<<(total + 255) / 256, 256, 0, stream>>>(xw, deg, bias, out, total);
    gcn_edge_scatter<<<(E * 32 + 255) / 256, 256, 0, stream>>>(src, dst, xw, deg, out, E);
    gcn_prelu<<<(total + 255) / 256, 256, 0, stream>>>(out, alpha, total);
}